// ShallowGMMConvNet_16561393893737
// MI455X (gfx1250) — compile-verified
//
#include <hip/hip_runtime.h>
#include <math.h>

typedef __attribute__((ext_vector_type(2))) float v2f;
typedef __attribute__((ext_vector_type(8))) float v8f;

#define NODES   100000
#define EDGES   1600000
#define KMIX    15
#define PITCH   112            // padded row stride (>= all channel counts, multiple of 16)
#define BSLOT   (PITCH*PITCH)  // one padded B matrix slot
#define SIG_EPS 1e-15f
#define ROWT    (NODES/16)     // 6250 row tiles
#define WPB     8              // waves (row strips) per block

// ---------------------------------------------------------------- utilities
__global__ void fill_kernel(float* __restrict__ p, long n, float v) {
    long i = (long)blockIdx.x * blockDim.x + threadIdx.x;
    long stride = (long)gridDim.x * blockDim.x;
    for (; i < n; i += stride) p[i] = v;
}

__global__ void copyin_kernel(const float* __restrict__ x, float* __restrict__ dst, int C) {
    long i = (long)blockIdx.x * blockDim.x + threadIdx.x;
    long n = (long)NODES * C;
    if (i >= n) return;
    int r = (int)(i / C), c = (int)(i % C);
    dst[(size_t)r * PITCH + c] = x[i];
}

__global__ void deg_kernel(const int* __restrict__ ei, float* __restrict__ deg) {
    int e = blockIdx.x * blockDim.x + threadIdx.x;
    if (e < EDGES) atomicAdd(&deg[ei[EDGES + e]], 1.0f);
}

__global__ void invdeg_kernel(const float* __restrict__ deg, float* __restrict__ inv) {
    int i = blockIdx.x * blockDim.x + threadIdx.x;
    if (i < NODES) inv[i] = 1.0f / fmaxf(deg[i], 1.0f);
}

// pack root (slot 0) and g_k (slots 1..15) into zero-padded 112x112 slots
__global__ void bprep_kernel(const float* __restrict__ g, const float* __restrict__ root,
                             float* __restrict__ B, int Cin, int Cout) {
    long i = (long)blockIdx.x * blockDim.x + threadIdx.x;
    long n = (long)(KMIX + 1) * BSLOT;
    if (i >= n) return;
    int slot = (int)(i / BSLOT);
    int rem  = (int)(i % BSLOT);
    int c = rem / PITCH, o = rem % PITCH;
    float v = 0.0f;
    if (c < Cin && o < Cout)
        v = (slot == 0) ? root[(size_t)c * Cout + o]
                        : g[((size_t)c * KMIX + (slot - 1)) * Cout + o];
    B[i] = v;
}

// ------------------------------------------------- WMMA f32 GEMM  C = A @ B
// A: [NODES x Kred] pitch PITCH (zero padded), B: [Kred x NT*16] pitch PITCH,
// C: [NODES x NT*16] pitch PITCH.
// Block = 8 waves; each wave owns one 16-row strip and ALL NT column tiles.
// B panel staged in LDS pair-interleaved: element (r,c) at ((r>>1)*NT16+c)*2+(r&1),
// so each lane's (K, K+1) B pair is one aligned ds_load_b64 feeding the WMMA.
template <int NT>
__global__ void gemm16_kernel(const float* __restrict__ A, const float* __restrict__ B,
                              float* __restrict__ C, int Kred) {
    constexpr int NT16 = NT * 16;
    __shared__ float bsh[PITCH * PITCH];   // >= Kred*NT16 for all layer shapes

    for (int i = threadIdx.x; i < Kred * NT16; i += blockDim.x) {
        int r = i / NT16, c = i - r * NT16;
        bsh[((r >> 1) * NT16 + c) * 2 + (r & 1)] = B[(size_t)r * PITCH + c];
    }
    __syncthreads();

    int lane = threadIdx.x & 31;
    int rowTile = blockIdx.x * WPB + (threadIdx.x >> 5);
    if (rowTile >= ROWT) return;           // wave-uniform exit (EXEC stays all-1s)

    int half  = lane >> 4;                 // 0: K=0,1   1: K=2,3
    int l16   = lane & 15;
    int khalf = half * 2;

    const float* arow = A + (size_t)(rowTile * 16 + l16) * PITCH + khalf;
    const float* bbase = bsh + (half * NT16 + l16) * 2;

    v8f acc[NT];
#pragma unroll
    for (int t = 0; t < NT; ++t) acc[t] = (v8f){0.f,0.f,0.f,0.f,0.f,0.f,0.f,0.f};

    for (int kk = 0; kk < Kred; kk += 4) {
        v2f a;
        a.x = arow[kk];
        a.y = arow[kk + 1];
        const float* b0 = bbase + kk * NT16;   // pair-row (kk/2) base for this lane
#pragma unroll
        for (int t = 0; t < NT; ++t) {
            v2f b = *(const v2f*)(b0 + t * 32);
            acc[t] = __builtin_amdgcn_wmma_f32_16x16x4_f32(false, a, false, b,
                                                           (short)0, acc[t], false, false);
        }
    }

    float* crow = C + (size_t)(rowTile * 16 + half * 8) * PITCH + l16;
#pragma unroll
    for (int t = 0; t < NT; ++t)
#pragma unroll
        for (int r = 0; r < 8; ++r)
            crow[(size_t)r * PITCH + t * 16] = acc[t][r];
}

// ------------------------------------------- edge scatter for one mixture k
// agg[tgt,:] += exp(-.5*||(u-mu_k)/sigma_k||^2) * invdeg[tgt] * P[src,:]
__global__ void scatter_kernel(const int* __restrict__ ei, const float* __restrict__ ea,
                               const float* __restrict__ mu, const float* __restrict__ sig,
                               const float* __restrict__ invdeg, const float* __restrict__ P,
                               float* __restrict__ agg, int k, int Cout) {
    int lane = threadIdx.x & 31;
    int wid  = (blockIdx.x * blockDim.x + threadIdx.x) >> 5;
    int nw   = (gridDim.x * blockDim.x) >> 5;
    float m0 = mu[k * 3 + 0], m1 = mu[k * 3 + 1], m2 = mu[k * 3 + 2];
    float s0 = sig[k * 3 + 0], s1 = sig[k * 3 + 1], s2 = sig[k * 3 + 2];
    s0 = 1.0f / (s0 * s0 + SIG_EPS);
    s1 = 1.0f / (s1 * s1 + SIG_EPS);
    s2 = 1.0f / (s2 * s2 + SIG_EPS);
    for (int e = wid; e < EDGES; e += nw) {
        int s = ei[e];
        int t = ei[EDGES + e];
        float d0 = ea[(size_t)e * 3 + 0] - m0;
        float d1 = ea[(size_t)e * 3 + 1] - m1;
        float d2 = ea[(size_t)e * 3 + 2] - m2;
        float w = expf(-0.5f * (d0 * d0 * s0 + d1 * d1 * s1 + d2 * d2 * s2)) * invdeg[t];
        const float* Ps = P + (size_t)s * PITCH;
        float* at = agg + (size_t)t * PITCH;
        for (int c = lane; c < Cout; c += 32)
            atomicAdd(&at[c], w * Ps[c]);
    }
}

// ------------------------------------------------------------ epilogues / BN
__global__ void elu_bias_kernel(float* __restrict__ h, const float* __restrict__ bias, int C) {
    long i = (long)blockIdx.x * blockDim.x + threadIdx.x;
    long n = (long)NODES * C;
    if (i >= n) return;
    int r = (int)(i / C), c = (int)(i % C);
    float v = h[(size_t)r * PITCH + c] + bias[c];
    h[(size_t)r * PITCH + c] = (v > 0.0f) ? v : expm1f(v);
}

__global__ void stats_kernel(const float* __restrict__ h, float* __restrict__ st, int C) {
    int c = threadIdx.x;
    if (c >= C) return;
    float s = 0.0f, ss = 0.0f;
    for (int r = blockIdx.x; r < NODES; r += gridDim.x) {
        float v = h[(size_t)r * PITCH + c];
        s += v; ss += v * v;
    }
    atomicAdd(&st[c], s);
    atomicAdd(&st[PITCH + c], ss);
}

__global__ void bn_kernel(const float* __restrict__ h, const float* __restrict__ st,
                          const float* __restrict__ gamma, const float* __restrict__ beta,
                          float* __restrict__ out, int C) {
    long i = (long)blockIdx.x * blockDim.x + threadIdx.x;
    long n = (long)NODES * C;
    if (i >= n) return;
    int r = (int)(i / C), c = (int)(i % C);
    const float invN = 1.0f / (float)NODES;
    float m   = st[c] * invN;
    float var = st[PITCH + c] * invN - m * m;
    float v   = h[(size_t)r * PITCH + c];
    out[(size_t)r * PITCH + c] = gamma[c] * (v - m) * rsqrtf(var + 1e-5f) + beta[c];
}

__global__ void final_kernel(const float* __restrict__ h, const float* __restrict__ bias,
                             float* __restrict__ out, int C) {
    long i = (long)blockIdx.x * blockDim.x + threadIdx.x;
    long n = (long)NODES * C;
    if (i >= n) return;
    int r = (int)(i / C), c = (int)(i % C);
    out[i] = h[(size_t)r * PITCH + c] + bias[c];
}

// ------------------------------------------------------------------ driver
static void launch_gemm(int colTiles, const float* A, const float* B, float* C,
                        int Kred, hipStream_t stream) {
    int blocksN = (ROWT + WPB - 1) / WPB;
    switch (colTiles) {
        case 4: gemm16_kernel<4><<<blocksN, 32 * WPB, 0, stream>>>(A, B, C, Kred); break;
        case 5: gemm16_kernel<5><<<blocksN, 32 * WPB, 0, stream>>>(A, B, C, Kred); break;
        default: gemm16_kernel<7><<<blocksN, 32 * WPB, 0, stream>>>(A, B, C, Kred); break;
    }
}

extern "C" void kernel_launch(void* const* d_in, const int* in_sizes, int n_in,
                              void* d_out, int out_size, void* d_ws, size_t ws_size,
                              hipStream_t stream) {
    const float* x  = (const float*)d_in[0];
    const int*   ei = (const int*)d_in[1];
    const float* ea = (const float*)d_in[2];

    const float *g[4], *mu[4], *sg[4], *rt[4], *bs[4], *gm[3], *bt[3];
    int base = 3;
    for (int l = 0; l < 4; ++l) {
        g[l]  = (const float*)d_in[base + 0];
        mu[l] = (const float*)d_in[base + 1];
        sg[l] = (const float*)d_in[base + 2];
        rt[l] = (const float*)d_in[base + 3];
        bs[l] = (const float*)d_in[base + 4];
        if (l < 3) { gm[l] = (const float*)d_in[base + 5]; bt[l] = (const float*)d_in[base + 6]; base += 7; }
        else base += 5;
    }
    const int Cin[4]  = {50, 75, 100, 75};
    const int Cout[4] = {75, 100, 75, 50};

    // workspace layout (floats)
    float* w   = (float*)d_ws;
    float* xA  = w;                                   // NODES*PITCH
    float* xB  = xA  + (size_t)NODES * PITCH;         // NODES*PITCH
    float* P   = xB  + (size_t)NODES * PITCH;         // NODES*PITCH
    float* agg = P   + (size_t)NODES * PITCH;         // NODES*PITCH
    float* deg = agg + (size_t)NODES * PITCH;         // NODES
    float* idg = deg + NODES;                         // NODES
    float* Bp  = idg + NODES;                         // 16*BSLOT
    float* st  = Bp  + (size_t)(KMIX + 1) * BSLOT;    // 2*PITCH

    auto blocks = [](long n, int bs_) { return (int)((n + bs_ - 1) / bs_); };

    // init: zero ping-pong node buffers (establishes zero padding), deg
    fill_kernel<<<2048, 256, 0, stream>>>(xA, (long)2 * NODES * PITCH, 0.0f);
    fill_kernel<<<blocks(NODES, 256), 256, 0, stream>>>(deg, NODES, 0.0f);
    copyin_kernel<<<blocks((long)NODES * 50, 256), 256, 0, stream>>>(x, xA, 50);
    deg_kernel<<<blocks(EDGES, 256), 256, 0, stream>>>(ei, deg);
    invdeg_kernel<<<blocks(NODES, 256), 256, 0, stream>>>(deg, idg);

    float* cur = xA;
    float* nxt = xB;
    for (int l = 0; l < 4; ++l) {
        int ci = Cin[l], co = Cout[l];
        int Kred = (ci + 3) & ~3;                 // 52,76,100,76
        int colTiles = (co + 15) / 16;            // 5,7,5,4

        // pack root + all g_k into padded B slots
        bprep_kernel<<<blocks((long)(KMIX + 1) * BSLOT, 256), 256, 0, stream>>>(g[l], rt[l], Bp, ci, co);

        // agg = x @ root   (v_wmma_f32_16x16x4_f32, B panel from LDS)
        launch_gemm(colTiles, cur, Bp, agg, Kred, stream);

        // per mixture component: P = x @ g_k, then edge scatter into agg
        for (int k = 0; k < KMIX; ++k) {
            launch_gemm(colTiles, cur, Bp + (size_t)(k + 1) * BSLOT, P, Kred, stream);
            scatter_kernel<<<4096, 256, 0, stream>>>(ei, ea, mu[l], sg[l], idg, P, agg, k, co);
        }

        if (l < 3) {
            elu_bias_kernel<<<blocks((long)NODES * co, 256), 256, 0, stream>>>(agg, bs[l], co);
            fill_kernel<<<1, 256, 0, stream>>>(st, 2 * PITCH, 0.0f);
            stats_kernel<<<1024, 128, 0, stream>>>(agg, st, co);
            bn_kernel<<<blocks((long)NODES * co, 256), 256, 0, stream>>>(agg, st, gm[l], bt[l], nxt, co);
            float* t = cur; cur = nxt; nxt = t;
        } else {
            final_kernel<<<blocks((long)NODES * co, 256), 256, 0, stream>>>(agg, bs[l], (float*)d_out, co);
        }
    }
}